// GTLayer_18210661335372
// MI455X (gfx1250) — compile-verified
//
#include <hip/hip_runtime.h>
#include <hip/hip_bf16.h>

// ---------------------------------------------------------------------------
// Graph-transformer layer for MI455X (gfx1250, wave32).
//
//  1) Node-level QKV projection with V_WMMA_F32_16X16X4_F32:
//     Q/K/V = embeds @ {qW,kW,vW} -> 3 x [N,128] in workspace (153.6 MB total,
//     fits the 192 MB L2, so edge-phase random gathers stay on-chip).
//     Embeds tile staged into LDS with GLOBAL_LOAD_ASYNC_TO_LDS_B128 (ASYNCcnt).
//  2) Edge pass A: wave-per-edge q.k per head, clip/exp, atomic denominator.
//  3) Edge pass B: normalize, write att, scatter att*V with f32 HW atomics.
// ---------------------------------------------------------------------------

typedef float v2f __attribute__((ext_vector_type(2)));
typedef float v8f __attribute__((ext_vector_type(8)));
typedef int   v4i __attribute__((vector_size(16)));   // matches builtin param

#define LATDIM 128
#define NHEAD  4

#define AS1 __attribute__((address_space(1)))
#define AS3 __attribute__((address_space(3)))

__device__ __forceinline__ void wait_asynccnt0() {
#if __has_builtin(__builtin_amdgcn_s_wait_asynccnt)
    __builtin_amdgcn_s_wait_asynccnt(0);
#else
    asm volatile("s_wait_asynccnt 0" ::: "memory");
#endif
}

// ---------------------------------------------------------------------------
// Kernel 1: Q/K/V = embeds @ {qW,kW,vW} using V_WMMA_F32_16X16X4_F32.
// Block = 256 threads = 8 waves; block owns a 16-node M-tile (LDS-staged);
// wave w owns output columns [16w,16w+16) for all three matrices.
//
// Fragment layouts per CDNA5 ISA 7.12.2:
//   A (16x4, MxK):  lane L: m = L&15; VGPR0 = A[m][2*(L>>4)], VGPR1 = +1
//   B (4x16, KxN):  mirrored: lane L: n = L&15; b.x = B[2*(L>>4)][n], b.y = +1
//   C/D (16x16):    VGPR r, lane L -> D[r + 8*(L>>4)][L&15]
// ---------------------------------------------------------------------------
__global__ __launch_bounds__(256) void qkv_gemm_wmma(
    const float* __restrict__ emb,
    const float* __restrict__ qW,
    const float* __restrict__ kW,
    const float* __restrict__ vW,
    float* __restrict__ Qo,
    float* __restrict__ Ko,
    float* __restrict__ Vo,
    int nNodes)
{
    __shared__ float tileA[16 * LATDIM];   // 8 KB

    const int  mbase    = blockIdx.x * 16;
    const bool fullTile = (mbase + 16 <= nNodes);

    // ---- Stage the 16x128 embeds tile into LDS ----------------------------
    if (fullTile) {
#if __has_builtin(__builtin_amdgcn_global_load_async_to_lds_b128)
        // CDNA5 async DMA: global -> LDS directly, tracked by ASYNCcnt.
        {
            const float4* src = (const float4*)(emb + (size_t)mbase * LATDIM);
            for (int i = threadIdx.x; i < (16 * LATDIM) / 4; i += blockDim.x) {
                AS1 v4i* gptr = (AS1 v4i*)(unsigned long long)(const void*)(src + i);
                AS3 v4i* lptr = (AS3 v4i*)(unsigned)(unsigned long long)(const void*)&tileA[i * 4];
                __builtin_amdgcn_global_load_async_to_lds_b128(gptr, lptr, 0, 0);
            }
            wait_asynccnt0();
        }
#else
        {
            const float4* src = (const float4*)(emb + (size_t)mbase * LATDIM);
            float4*       dst = (float4*)tileA;
            for (int i = threadIdx.x; i < (16 * LATDIM) / 4; i += blockDim.x)
                dst[i] = src[i];
        }
#endif
    } else {
        const float4* src = (const float4*)(emb + (size_t)mbase * LATDIM);
        float4*       dst = (float4*)tileA;
        for (int i = threadIdx.x; i < (16 * LATDIM) / 4; i += blockDim.x) {
            int row = (i * 4) / LATDIM;
            dst[i] = (mbase + row < nNodes) ? src[i]
                                            : make_float4(0.f, 0.f, 0.f, 0.f);
        }
    }
    __syncthreads();

    const int wave = threadIdx.x >> 5;   // 0..7
    const int lane = threadIdx.x & 31;
    const int n0   = wave * 16;
    const int m    = lane & 15;
    const int kg   = lane >> 4;          // 0 or 1
    const int n    = lane & 15;

    const float* Ws[3] = {qW, kW, vW};
    float*       Os[3] = {Qo, Ko, Vo};

#pragma unroll
    for (int mat = 0; mat < 3; ++mat) {
        const float* __restrict__ W = Ws[mat];
        v8f acc = {0.f, 0.f, 0.f, 0.f, 0.f, 0.f, 0.f, 0.f};

#pragma unroll 4
        for (int k0 = 0; k0 < LATDIM; k0 += 4) {
            const int ka = k0 + 2 * kg;
            v2f a;
            a.x = tileA[m * LATDIM + ka];        // ds_load_b64 pair
            a.y = tileA[m * LATDIM + ka + 1];
            v2f b;
            b.x = W[(size_t)ka * LATDIM + n0 + n];
            b.y = W[(size_t)(ka + 1) * LATDIM + n0 + n];
            // 8-arg f32 WMMA: (neg_a, A, neg_b, B, c_mod, C, reuse_a, reuse_b)
            acc = __builtin_amdgcn_wmma_f32_16x16x4_f32(
                false, a, false, b, (short)0, acc, false, false);
        }

        float* __restrict__ O = Os[mat] + (size_t)mbase * LATDIM + n0 + n;
        if (fullTile) {
            // Guard-free store clause (always taken: N % 16 == 0).
#pragma unroll
            for (int r = 0; r < 8; ++r)
                O[(size_t)(r + 8 * kg) * LATDIM] = acc[r];
        } else {
#pragma unroll
            for (int r = 0; r < 8; ++r)
                if (mbase + r + 8 * kg < nNodes)
                    O[(size_t)(r + 8 * kg) * LATDIM] = acc[r];
        }
    }
}

// ---------------------------------------------------------------------------
// Kernel 2 (edge pass A): att_h = exp(clip(q.k)); accumulate denominator.
// Wave per edge; lane l holds dims [4l, 4l+4); head h = l>>3.
// ---------------------------------------------------------------------------
__global__ __launch_bounds__(256) void edge_att_kernel(
    const float* __restrict__ Q,
    const float* __restrict__ K,
    const int*   __restrict__ rows,
    const int*   __restrict__ cols,
    float* __restrict__ attNorm,     // [N, 4], pre-zeroed
    float* __restrict__ attOut,      // [E, 4] (holds expAtt after this pass)
    int nEdges)
{
    const int edge = (blockIdx.x * blockDim.x + threadIdx.x) >> 5;
    const int lane = threadIdx.x & 31;
    if (edge >= nEdges) return;      // wave-uniform

    const int r = rows[edge];
    const int c = cols[edge];

    const float4 q = ((const float4*)(Q + (size_t)r * LATDIM))[lane];
    const float4 k = ((const float4*)(K + (size_t)c * LATDIM))[lane];

    float p = q.x * k.x + q.y * k.y + q.z * k.z + q.w * k.w;
    // Reduce within the 8-lane head group (xor masks < 8 stay in-group).
    p += __shfl_xor(p, 1, 32);
    p += __shfl_xor(p, 2, 32);
    p += __shfl_xor(p, 4, 32);

    p = fminf(fmaxf(p, -10.0f), 10.0f);
    const float e = __expf(p);

    if ((lane & 7) == 0) {
        const int h = lane >> 3;
        attOut[(size_t)edge * NHEAD + h] = e;
        unsafeAtomicAdd(&attNorm[(size_t)r * NHEAD + h], e);
    }
}

// ---------------------------------------------------------------------------
// Kernel 3 (edge pass B): normalize, write att, scatter att*V into resEmbeds.
// ---------------------------------------------------------------------------
__global__ __launch_bounds__(256) void edge_agg_kernel(
    const float* __restrict__ V,
    const int*   __restrict__ rows,
    const int*   __restrict__ cols,
    const float* __restrict__ attNorm,
    float* __restrict__ attOut,      // in: expAtt, out: normalized att
    float* __restrict__ res,         // [N, 128], pre-zeroed
    int nEdges)
{
    const int edge = (blockIdx.x * blockDim.x + threadIdx.x) >> 5;
    const int lane = threadIdx.x & 31;
    if (edge >= nEdges) return;      // wave-uniform

    const int r = rows[edge];
    const int c = cols[edge];
    const int h = lane >> 3;

    const float eAtt = attOut[(size_t)edge * NHEAD + h];
    const float a    = eAtt / (attNorm[(size_t)r * NHEAD + h] + 1e-8f);

    if ((lane & 7) == 0)
        attOut[(size_t)edge * NHEAD + h] = a;

    const float4 v = ((const float4*)(V + (size_t)c * LATDIM))[lane];
    float* dst = res + (size_t)r * LATDIM + lane * 4;
    unsafeAtomicAdd(dst + 0, a * v.x);
    unsafeAtomicAdd(dst + 1, a * v.y);
    unsafeAtomicAdd(dst + 2, a * v.z);
    unsafeAtomicAdd(dst + 3, a * v.w);
}

// ---------------------------------------------------------------------------
// Host launcher. Inputs: embeds, qW, kW, vW, rows, cols, n_nodes.
// Output: [resEmbeds (N*128) | att (E*4)] flat f32.
// Workspace: Q | K | V | attNorm  (~155 MB).
// ---------------------------------------------------------------------------
extern "C" void kernel_launch(void* const* d_in, const int* in_sizes, int n_in,
                              void* d_out, int out_size, void* d_ws, size_t ws_size,
                              hipStream_t stream)
{
    const float* emb  = (const float*)d_in[0];
    const float* qW   = (const float*)d_in[1];
    const float* kW   = (const float*)d_in[2];
    const float* vW   = (const float*)d_in[3];
    const int*   rows = (const int*)d_in[4];
    const int*   cols = (const int*)d_in[5];

    const int N = in_sizes[0] / LATDIM;   // 100000
    const int E = in_sizes[4];            // 600000

    float* Q       = (float*)d_ws;
    float* K       = Q + (size_t)N * LATDIM;
    float* V       = K + (size_t)N * LATDIM;
    float* attNorm = V + (size_t)N * LATDIM;

    float* res    = (float*)d_out;                 // [N, 128]
    float* attOut = res + (size_t)N * LATDIM;      // [E, 4]

    // Zero the scatter targets (stream-ordered; graph-capture safe).
    (void)hipMemsetAsync(res,     0, (size_t)N * LATDIM * sizeof(float), stream);
    (void)hipMemsetAsync(attNorm, 0, (size_t)N * NHEAD  * sizeof(float), stream);

    // 1) QKV projection (WMMA f32).
    const int mtiles = (N + 15) / 16;
    qkv_gemm_wmma<<<mtiles, 256, 0, stream>>>(emb, qW, kW, vW, Q, K, V, N);

    // 2) + 3) Edge phases: one wave32 per edge, 8 edges per 256-thread block.
    const int edgeBlocks = (E + 7) / 8;
    edge_att_kernel<<<edgeBlocks, 256, 0, stream>>>(Q, K, rows, cols,
                                                    attNorm, attOut, E);
    edge_agg_kernel<<<edgeBlocks, 256, 0, stream>>>(V, rows, cols,
                                                    attNorm, attOut, res, E);
}